// LieAction_91156385891024
// MI455X (gfx1250) — compile-verified
//
#include <hip/hip_runtime.h>
#include <hip/hip_bf16.h>

// ---------------------------------------------------------------------------
// LieAction: out[b,k] = expm(act_params[act[b],k] * (L_k - L_k^T)) @ G[b,k]
// B=16384, K=64, M=6, NUM_ACTIONS=16.
//
// Only 16*64 = 1024 distinct matrix exponentials exist -> precompute.
// Main kernel = memory-bound batched 6x6 matmul (302 MB traffic ~ 13us floor
// at 23.3 TB/s), mapped onto v_wmma_f32_16x16x4_f32: two 6x6 blocks packed
// block-diagonally per 16x16 tile, K=12 -> 3 chained WMMAs. Staging uses the
// gfx1250 async LDS engine (global_load_async_to_lds_b128 /
// global_store_async_from_lds_b128, ASYNCcnt).
// ---------------------------------------------------------------------------

#define NB 16384       // batch
#define NK 64          // subgroups
#define NA 16          // actions
#define MM 6           // block dim
#define BLK36 36       // MM*MM

typedef __attribute__((ext_vector_type(2))) float v2f;
typedef __attribute__((ext_vector_type(8))) float v8f;

// ---------------------------------------------------------------------------
// Kernel 1: E[a][k] = expm(act_params[a,k] * (L_k - L_k^T)), 4th-order Taylor.
// act_params ~ N(0,1)*1e-3 => ||A|| ~ 1e-2, 4th-order error ~1e-10 << fp32 eps.
// ---------------------------------------------------------------------------
__global__ void lie_expm_table(const float* __restrict__ act_params,   // [16,64]
                               const float* __restrict__ basis,        // [64,6,6]
                               float* __restrict__ E)                  // [16,64,36]
{
    int t = blockIdx.x * blockDim.x + threadIdx.x;
    if (t >= NA * NK) return;
    int a = t >> 6, k = t & 63;
    float s = act_params[a * NK + k];
    const float* L = basis + k * BLK36;

    float A[BLK36], P[BLK36], T[BLK36];
    #pragma unroll
    for (int i = 0; i < MM; ++i)
        #pragma unroll
        for (int j = 0; j < MM; ++j)
            A[i * MM + j] = s * (L[i * MM + j] - L[j * MM + i]);

    // Horner: E = I + A(I + A/2 (I + A/3 (I + A/4)))
    #pragma unroll
    for (int i = 0; i < BLK36; ++i) P[i] = 0.25f * A[i];
    #pragma unroll
    for (int i = 0; i < MM; ++i) P[i * 7] += 1.0f;

    #pragma unroll
    for (int d = 3; d >= 1; --d) {
        float inv = 1.0f / (float)d;
        #pragma unroll
        for (int i = 0; i < MM; ++i)
            #pragma unroll
            for (int j = 0; j < MM; ++j) {
                float acc = 0.0f;
                #pragma unroll
                for (int m = 0; m < MM; ++m)
                    acc = __builtin_fmaf(A[i * MM + m], P[m * MM + j], acc);
                T[i * MM + j] = inv * acc;
            }
        #pragma unroll
        for (int i = 0; i < BLK36; ++i) P[i] = T[i];
        #pragma unroll
        for (int i = 0; i < MM; ++i) P[i * 7] += 1.0f;
    }

    #pragma unroll
    for (int i = 0; i < BLK36; ++i) E[t * BLK36 + i] = P[i];
}

// ---------------------------------------------------------------------------
// Kernel 2: pre-swizzle E into the WMMA A-fragment layout.
// Tile = (block 2kp, block 2kp+1) of one action: A(16x12) block-diag:
//   A[r,c] = T0[r,c]      (r<6,  c<6)
//   A[r,c] = T1[r-8,c-6]  (8<=r<14, 6<=c<12), else 0
// A-fragment (16x16x4 f32): lane (hl=lane>>4, l16=lane&15), VGPR j of step s
// holds A[l16, 4s + 2*hl + j]. 6 values + 2 pad per lane -> two b128 loads.
// TA layout: [a][kp][lane][8] = 512 KB (L2-resident forever).
// ---------------------------------------------------------------------------
__global__ void lie_build_ta(const float* __restrict__ E,   // [16,64,36]
                             float* __restrict__ TA)        // [16,32,32,8]
{
    int tid = blockIdx.x * blockDim.x + threadIdx.x;   // 16384 threads
    int lane = tid & 31;
    int kp   = (tid >> 5) & 31;
    int a    = tid >> 10;
    int hl = lane >> 4, l16 = lane & 15;

    float frag[8];
    #pragma unroll
    for (int s = 0; s < 3; ++s)
        #pragma unroll
        for (int j = 0; j < 2; ++j) {
            int c = 4 * s + 2 * hl + j;               // 0..11
            float v = 0.0f;
            if (c < 6) {
                if (l16 < 6)
                    v = E[(a * NK + 2 * kp) * BLK36 + l16 * MM + c];
            } else {
                int r2 = l16 - 8;
                if (r2 >= 0 && r2 < 6)
                    v = E[(a * NK + 2 * kp + 1) * BLK36 + r2 * MM + (c - 6)];
            }
            frag[2 * s + j] = v;
        }
    frag[6] = 0.0f; frag[7] = 0.0f;

    float4* dst = (float4*)(TA + (size_t)tid * 8);
    dst[0] = make_float4(frag[0], frag[1], frag[2], frag[3]);
    dst[1] = make_float4(frag[4], frag[5], frag[6], frag[7]);
}

// ---------------------------------------------------------------------------
// Kernel 3: streaming apply. Each wave owns 32 consecutive 6x6 blocks
// (all within one batch row b = w>>1 -> single action per wave).
// async b128 loads -> LDS, 16 tiles x 3 WMMAs (fully unrolled), D written
// back over the consumed G region, async b128 stores LDS -> memory.
// Invalid B-fragment lanes feed only never-stored / zero-A products, so
// they load clamped don't-care addresses: zero predication on the hot path.
// ---------------------------------------------------------------------------
__global__ __launch_bounds__(256) void lie_apply(
    const float4* __restrict__ gf4,   // group_feats as float4
    const int*    __restrict__ act,   // [B]
    const float*  __restrict__ TA,    // [16,32,32,8]
    float4*       __restrict__ out4)  // same shape as gf4
{
    __shared__ __align__(16) float lds[8][1160];   // 4.6KB staged + pad, per wave

    const int lane = threadIdx.x & 31;
    const int wv   = threadIdx.x >> 5;
    const int w    = blockIdx.x * 8 + wv;          // global wave id (32768 total)
    const int hl   = lane >> 4;
    const int l16  = lane & 15;
    float* g = lds[wv];

    // LDS byte offset of this wave's staging region (module LDS starts at 0)
    const unsigned goff =
        (unsigned)((unsigned long long)(g) - (unsigned long long)(&lds[0][0]));

    // ---- stage 32 blocks (288 float4) straight into LDS (ASYNCcnt) ----
    const size_t gbase = (size_t)w * 288;
    #pragma unroll
    for (int i = 0; i < 9; ++i) {
        unsigned dst = goff + (unsigned)((i * 32 + lane) * 16);
        const float4* src = gf4 + (gbase + i * 32 + lane);
        asm volatile("global_load_async_to_lds_b128 %0, %1, off"
                     :: "v"(dst), "v"((unsigned long long)src)
                     : "memory");
    }

    const int a       = act[w >> 1];               // uniform across the wave
    const int kp_base = (w & 1) * 16;

    // D write-back: cfrag[v] = D[v + 8*hl, l16]
    //   block0 (hl==0): row v, col l16   (l16<6)
    //   block1 (hl==1): row v, col l16-8 (8<=l16<14)
    const bool dpred = hl ? (l16 >= 8 && l16 < 14) : (l16 < 6);
    const int  dcol  = hl ? (l16 - 8) : l16;

    // Branchless B-fragment offsets within a 72-float tile region.
    // frag f = 2*s + j holds B[c, l16], c = 4s + 2*hl + j.  Invalid lanes are
    // don't-care -> clamp the column and load unconditionally.
    int boff[6];
    #pragma unroll
    for (int f = 0; f < 6; ++f) {
        int c   = 4 * (f >> 1) + 2 * hl + (f & 1); // 0..11
        int blk = (c >= 6) ? 1 : 0;
        int row = c - 6 * blk;
        int col = blk ? (l16 - 8) : l16;
        col = col < 0 ? 0 : (col > 5 ? 5 : col);   // clamp: don't-care data
        boff[f] = blk * 36 + row * MM + col;
    }

    asm volatile("s_wait_asynccnt 0x0" ::: "memory");  // staged G visible

    #pragma unroll
    for (int t = 0; t < 16; ++t) {
        const int kp = kp_base + t;

        // A fragments: two b128 loads from the pre-swizzled (L2-hot) table
        const float4* ta = (const float4*)(TA +
            ((((size_t)a * 32 + kp) * 32 + lane) * 8));
        float4 tA = ta[0];
        float4 tB = ta[1];

        // B fragments: 6 unconditional ds_load_b32 (immediate offset t*288)
        float bf[6];
        #pragma unroll
        for (int f = 0; f < 6; ++f)
            bf[f] = g[t * 72 + boff[f]];

        v2f a0 = { tA.x, tA.y }, a1 = { tA.z, tA.w }, a2 = { tB.x, tB.y };
        v2f b0 = { bf[0], bf[1] }, b1 = { bf[2], bf[3] }, b2 = { bf[4], bf[5] };

        v8f c8 = { 0.f, 0.f, 0.f, 0.f, 0.f, 0.f, 0.f, 0.f };
        c8 = __builtin_amdgcn_wmma_f32_16x16x4_f32(false, a0, false, b0,
                                                   (short)0, c8, false, false);
        c8 = __builtin_amdgcn_wmma_f32_16x16x4_f32(false, a1, false, b1,
                                                   (short)0, c8, false, false);
        c8 = __builtin_amdgcn_wmma_f32_16x16x4_f32(false, a2, false, b2,
                                                   (short)0, c8, false, false);

        // write D over the consumed G region (rows v=0..5, stride 6)
        if (dpred) {
            float* ob = &g[t * 72 + (hl ? 36 : 0) + dcol];
            ob[0 * 6] = c8[0];
            ob[1 * 6] = c8[1];
            ob[2 * 6] = c8[2];
            ob[3 * 6] = c8[3];
            ob[4 * 6] = c8[4];
            ob[5 * 6] = c8[5];
        }
    }

    // ---- stream results out: async LDS -> memory b128 stores ----
    asm volatile("s_wait_dscnt 0x0" ::: "memory");     // D writes visible
    #pragma unroll
    for (int i = 0; i < 9; ++i) {
        unsigned src = goff + (unsigned)((i * 32 + lane) * 16);
        float4* dst = out4 + (gbase + i * 32 + lane);
        asm volatile("global_store_async_from_lds_b128 %0, %1, off"
                     :: "v"((unsigned long long)dst), "v"(src)
                     : "memory");
    }
    // s_endpgm performs an implicit wait-idle, covering outstanding ASYNCcnt
}

// ---------------------------------------------------------------------------
extern "C" void kernel_launch(void* const* d_in, const int* in_sizes, int n_in,
                              void* d_out, int out_size, void* d_ws, size_t ws_size,
                              hipStream_t stream) {
    const float* group_feats = (const float*)d_in[0];   // [16384, 64*36] f32
    const int*   act         = (const int*)d_in[1];     // [16384] i32
    const float* act_params  = (const float*)d_in[2];   // [16, 64] f32
    const float* basis       = (const float*)d_in[3];   // [64, 6, 6] f32

    float* E  = (float*)d_ws;                 // 16*64*36   = 36864 floats
    float* TA = E + (NA * NK * BLK36);        // 16*32*32*8 = 131072 floats

    // Phase 1: 1024 matrix exponentials
    lie_expm_table<<<4, 256, 0, stream>>>(act_params, basis, E);
    // Phase 2: swizzle into WMMA A-fragment layout
    lie_build_ta<<<64, 256, 0, stream>>>(E, TA);
    // Phase 3: streaming batched 6x6 matmul via WMMA
    // B*K / (32 blocks per wave) = 32768 waves = 4096 workgroups of 8 waves
    lie_apply<<<4096, 256, 0, stream>>>((const float4*)group_feats, act, TA,
                                        (float4*)d_out);
}